// RepLKBlock_2525440770744
// MI455X (gfx1250) — compile-verified
//
#include <hip/hip_runtime.h>
#include <hip/hip_bf16.h>
#include <math.h>

typedef float v2f __attribute__((ext_vector_type(2)));
typedef float v8f __attribute__((ext_vector_type(8)));

#define EPS 1e-5f
constexpr int C_  = 64;
constexpr int D_  = 48;
constexpr int P_  = D_*D_*D_;      // 110592
constexpr int B_  = 2;
constexpr int NP_ = B_*P_;         // 221184 positions total
constexpr long long NE_ = (long long)B_*C_*P_;  // 14155776 elements / tensor

// ---- fp32 WMMA wrapper: D = A(16x4) * B(4x16) + C  (V_WMMA_F32_16X16X4_F32) ----
static __device__ inline v8f wmma4(v2f a, v2f b, v8f c) {
  return __builtin_amdgcn_wmma_f32_16x16x4_f32(false, a, false, b, (short)0, c, false, false);
}

static __device__ inline v8f vzero8() {
  v8f z;
  #pragma unroll
  for (int m = 0; m < 8; ++m) z[m] = 0.f;
  return z;
}

// =============================== utility kernels ===============================
__global__ void k_zero(float* p, int n) {
  int i = blockIdx.x*blockDim.x + threadIdx.x;
  if (i < n) p[i] = 0.f;
}

// per-channel sum & sumsq of a (B,C,P) tensor -> st[c], st[64+c]
__global__ void k_stats(const float* __restrict__ X, float* __restrict__ st) {
  __shared__ float red[2];
  int s = blockIdx.x, c = blockIdx.y, n = blockIdx.z, tid = threadIdx.x;
  if (tid < 2) red[tid] = 0.f;
  __syncthreads();
  const float* p = X + (long long)(n*C_ + c)*P_ + s*(P_/16);
  float sm = 0.f, sq = 0.f;
  for (int i = tid; i < P_/16; i += 256) { float v = p[i]; sm += v; sq += v*v; }
  atomicAdd(&red[0], sm); atomicAdd(&red[1], sq);
  __syncthreads();
  if (tid == 0) { atomicAdd(&st[c], red[0]); atomicAdd(&st[C_+c], red[1]); }
}

// O = a[c]*Y + b[c] + X  (+ stats of O)
__global__ void k_resid_stats(const float* __restrict__ Y, const float* __restrict__ X,
                              const float* __restrict__ A, const float* __restrict__ Cc,
                              float* __restrict__ O, float* __restrict__ st) {
  __shared__ float red[2];
  int s = blockIdx.x, c = blockIdx.y, n = blockIdx.z, tid = threadIdx.x;
  if (tid < 2) red[tid] = 0.f;
  __syncthreads();
  long long base = (long long)(n*C_ + c)*P_ + s*(P_/16);
  float a = A[c], cc = Cc[c], sm = 0.f, sq = 0.f;
  for (int i = tid; i < P_/16; i += 256) {
    float v = a*Y[base+i] + cc + X[base+i];
    O[base+i] = v; sm += v; sq += v*v;
  }
  atomicAdd(&red[0], sm); atomicAdd(&red[1], sq);
  __syncthreads();
  if (tid == 0) { atomicAdd(&st[c], red[0]); atomicAdd(&st[C_+c], red[1]); }
}

__global__ void k_resid_out(const float* __restrict__ Y, const float* __restrict__ X,
                            const float* __restrict__ A, const float* __restrict__ Cc,
                            float* __restrict__ O) {
  int s = blockIdx.x, c = blockIdx.y, n = blockIdx.z, tid = threadIdx.x;
  long long base = (long long)(n*C_ + c)*P_ + s*(P_/16);
  float a = A[c], cc = Cc[c];
  for (int i = tid; i < P_/16; i += 256)
    O[base+i] = a*Y[base+i] + cc + X[base+i];
}

// stats -> per-channel affine:  A = g/sqrt(v+eps), C = b - m*A
__global__ void k_affine(const float* __restrict__ st, const float* __restrict__ g,
                         const float* __restrict__ b, float* __restrict__ A,
                         float* __restrict__ Cc, int nch) {
  int c = blockIdx.x*blockDim.x + threadIdx.x;
  if (c < nch) {
    float inv = 1.f/(float)NP_;
    float m = st[c]*inv, v = st[nch+c]*inv - m*m;
    float a = g[c]*rsqrtf(v + EPS);
    A[c] = a; Cc[c] = b[c] - m*a;
  }
}

// =============================== finalize/fold kernels ===============================
// fold per_bn into pc weights (store transposed [k][n]) + bias; also transpose f2_w
__global__ void k_fin1(const float* __restrict__ st_x, const float* __restrict__ g1,
                       const float* __restrict__ b1, const float* __restrict__ pcw,
                       const float* __restrict__ pcb, const float* __restrict__ f2w,
                       float* __restrict__ WTpc, float* __restrict__ biaspc,
                       float* __restrict__ W2T) {
  __shared__ float sA[64], sC[64];
  int tid = threadIdx.x;
  if (tid < 64) {
    float inv = 1.f/(float)NP_;
    float m = st_x[tid]*inv, v = st_x[64+tid]*inv - m*m;
    float a = g1[tid]*rsqrtf(v + EPS);
    sA[tid] = a; sC[tid] = b1[tid] - m*a;
  }
  __syncthreads();
  for (int idx = tid; idx < 4096; idx += 256) {
    int i = idx >> 6, o = idx & 63;          // WTpc[i][o] = pcw[o][i]*a1[i]
    WTpc[idx] = pcw[o*64+i]*sA[i];
  }
  if (tid < 64) {
    float b = pcb[tid];
    for (int i = 0; i < 64; ++i) b += pcw[tid*64+i]*sC[i];
    biaspc[tid] = b;
  }
  for (int idx = tid; idx < 16384; idx += 256) {
    int i = idx >> 6, o = idx & 63;          // W2T[i][o] = f2w[o][i]
    W2T[idx] = f2w[o*256+i];
  }
}

// compose lk_bn∘lk2 and sk_bn∘sk2 affines, fold into po weights (+bias)
__global__ void k_fin3(const float* __restrict__ st1, const float* __restrict__ st2,
                       const float* __restrict__ lkbng, const float* __restrict__ lk2g,
                       const float* __restrict__ lk2b, const float* __restrict__ skbng,
                       const float* __restrict__ sk2g, const float* __restrict__ sk2b,
                       const float* __restrict__ poww, const float* __restrict__ pob,
                       float* __restrict__ WAT, float* __restrict__ WBT,
                       float* __restrict__ biaspo) {
  __shared__ float G1[64], H1[64], G2[64], H2[64];
  int tid = threadIdx.x;
  if (tid < 64) {
    float inv = 1.f/(float)NP_;
    float m = st1[tid]*inv, v = st1[64+tid]*inv - m*m;
    float al = lkbng[tid]*rsqrtf(v + EPS);
    float s2 = lk2g[tid]*rsqrtf(al*al*v + EPS);   // mean of 1st BN output = its beta (cancels)
    float gg = s2*al; G1[tid] = gg; H1[tid] = lk2b[tid] - gg*m;
    m = st2[tid]*inv; v = st2[64+tid]*inv - m*m;
    al = skbng[tid]*rsqrtf(v + EPS);
    s2 = sk2g[tid]*rsqrtf(al*al*v + EPS);
    gg = s2*al; G2[tid] = gg; H2[tid] = sk2b[tid] - gg*m;
  }
  __syncthreads();
  for (int idx = tid; idx < 4096; idx += 256) {
    int i = idx >> 6, o = idx & 63;
    WAT[idx] = poww[o*64+i]*G1[i];
    WBT[idx] = poww[o*64+i]*G2[i];
  }
  if (tid < 64) {
    float b = pob[tid];
    for (int i = 0; i < 64; ++i) b += poww[tid*64+i]*(H1[i] + H2[i]);
    biaspo[tid] = b;
  }
}

// fold ffn_bn into f1 weights (store [k][n=256]) + bias
__global__ void k_fin5(const float* __restrict__ st_xr, const float* __restrict__ fg,
                       const float* __restrict__ fb, const float* __restrict__ f1w,
                       const float* __restrict__ f1b, float* __restrict__ W1T,
                       float* __restrict__ b1o) {
  __shared__ float sA[64], sC[64];
  int tid = threadIdx.x;
  if (tid < 64) {
    float inv = 1.f/(float)NP_;
    float m = st_xr[tid]*inv, v = st_xr[64+tid]*inv - m*m;
    float a = fg[tid]*rsqrtf(v + EPS);
    sA[tid] = a; sC[tid] = fb[tid] - m*a;
  }
  __syncthreads();
  for (int idx = tid; idx < 16384; idx += 256) {
    int i = idx >> 8, o = idx & 255;         // W1T[i][o] = f1w[o][i]*a_f[i]
    W1T[idx] = f1w[o*64+i]*sA[i];
  }
  { int o = tid; float b = f1b[o];
    for (int i = 0; i < 64; ++i) b += f1w[o*64+i]*sC[i];
    b1o[o] = b; }
}

// =============================== WMMA GEMM kernels ===============================
// hp[o,p] = sum_i WT[i][o]*x[i,p] + bias[o]; per-channel stats of hp
__global__ __launch_bounds__(256) void k_gemm64(
    const float* __restrict__ X, const float* __restrict__ WT,
    const float* __restrict__ bias, float* __restrict__ Y, float* __restrict__ st) {
  __shared__ float sW[64*65];                 // padded [k][n]
  __shared__ float ssum[64], ssq[64];
  int tid = threadIdx.x;
  for (int i = tid; i < 4096; i += 256) sW[(i>>6)*65 + (i&63)] = WT[i];
  if (tid < 64) { ssum[tid] = 0.f; ssq[tid] = 0.f; }
  __syncthreads();
  int wave = tid>>5, lane = tid&31, half = lane>>4, row = lane&15;
  int q = blockIdx.x*128 + wave*16;
  int n = q/P_, p = q - n*P_;
  const float* xb = X + (long long)n*C_*P_ + p + row;
  v8f acc[4];
  #pragma unroll
  for (int t = 0; t < 4; ++t) acc[t] = vzero8();
  for (int k0 = 0; k0 < 64; k0 += 4) {
    int ka = k0 + half*2;
    v2f A; A.x = xb[ka*P_]; A.y = xb[(ka+1)*P_];
    #pragma unroll
    for (int t = 0; t < 4; ++t) {
      int nn = t*16 + row;
      v2f Bv; Bv.x = sW[ka*65+nn]; Bv.y = sW[(ka+1)*65+nn];
      acc[t] = wmma4(A, Bv, acc[t]);
    }
  }
  #pragma unroll
  for (int t = 0; t < 4; ++t) {
    int ch = t*16 + row;
    float bs = bias[ch], sm = 0.f, sq = 0.f, tmp[8];
    float* op = Y + (long long)(n*C_+ch)*P_ + p + half*8;
    #pragma unroll
    for (int m = 0; m < 8; ++m) { float v = acc[t][m] + bs; tmp[m] = v; sm += v; sq += v*v; }
    *(float4*)op     = make_float4(tmp[0], tmp[1], tmp[2], tmp[3]);
    *(float4*)(op+4) = make_float4(tmp[4], tmp[5], tmp[6], tmp[7]);
    atomicAdd(&ssum[ch], sm); atomicAdd(&ssq[ch], sq);
  }
  __syncthreads();
  if (tid < 64) { atomicAdd(&st[tid], ssum[tid]); atomicAdd(&st[64+tid], ssq[tid]); }
}

// po[o,p] = WAT'*y1 + WBT'*y2 + bias ; stats of po
__global__ __launch_bounds__(256) void k_gemm64x2(
    const float* __restrict__ Xa, const float* __restrict__ Xb,
    const float* __restrict__ WTa, const float* __restrict__ WTb,
    const float* __restrict__ bias, float* __restrict__ Y, float* __restrict__ st) {
  __shared__ float sWa[64*65], sWb[64*65];
  __shared__ float ssum[64], ssq[64];
  int tid = threadIdx.x;
  for (int i = tid; i < 4096; i += 256) {
    int k = i>>6, o = i&63;
    sWa[k*65+o] = WTa[i]; sWb[k*65+o] = WTb[i];
  }
  if (tid < 64) { ssum[tid] = 0.f; ssq[tid] = 0.f; }
  __syncthreads();
  int wave = tid>>5, lane = tid&31, half = lane>>4, row = lane&15;
  int q = blockIdx.x*128 + wave*16;
  int n = q/P_, p = q - n*P_;
  const float* xa  = Xa + (long long)n*C_*P_ + p + row;
  const float* xbp = Xb + (long long)n*C_*P_ + p + row;
  v8f acc[4];
  #pragma unroll
  for (int t = 0; t < 4; ++t) acc[t] = vzero8();
  for (int k0 = 0; k0 < 64; k0 += 4) {
    int ka = k0 + half*2;
    v2f A; A.x = xa[ka*P_]; A.y = xa[(ka+1)*P_];
    #pragma unroll
    for (int t = 0; t < 4; ++t) {
      int nn = t*16 + row;
      v2f Bv; Bv.x = sWa[ka*65+nn]; Bv.y = sWa[(ka+1)*65+nn];
      acc[t] = wmma4(A, Bv, acc[t]);
    }
  }
  for (int k0 = 0; k0 < 64; k0 += 4) {
    int ka = k0 + half*2;
    v2f A; A.x = xbp[ka*P_]; A.y = xbp[(ka+1)*P_];
    #pragma unroll
    for (int t = 0; t < 4; ++t) {
      int nn = t*16 + row;
      v2f Bv; Bv.x = sWb[ka*65+nn]; Bv.y = sWb[(ka+1)*65+nn];
      acc[t] = wmma4(A, Bv, acc[t]);
    }
  }
  #pragma unroll
  for (int t = 0; t < 4; ++t) {
    int ch = t*16 + row;
    float bs = bias[ch], sm = 0.f, sq = 0.f, tmp[8];
    float* op = Y + (long long)(n*C_+ch)*P_ + p + half*8;
    #pragma unroll
    for (int m = 0; m < 8; ++m) { float v = acc[t][m] + bs; tmp[m] = v; sm += v; sq += v*v; }
    *(float4*)op     = make_float4(tmp[0], tmp[1], tmp[2], tmp[3]);
    *(float4*)(op+4) = make_float4(tmp[4], tmp[5], tmp[6], tmp[7]);
    atomicAdd(&ssum[ch], sm); atomicAdd(&ssq[ch], sq);
  }
  __syncthreads();
  if (tid < 64) { atomicAdd(&st[tid], ssum[tid]); atomicAdd(&st[64+tid], ssq[tid]); }
}

// stats-only pass over t = W1T'*xr + b1 (256 channels, split in halves over grid.y)
__global__ __launch_bounds__(256) void k_gemm256_stats(
    const float* __restrict__ X, const float* __restrict__ W1T,
    const float* __restrict__ b1, float* __restrict__ st) {
  __shared__ float ssum[128], ssq[128];
  int tid = threadIdx.x, hb = blockIdx.y;
  if (tid < 128) { ssum[tid] = 0.f; ssq[tid] = 0.f; }
  __syncthreads();
  int wave = tid>>5, lane = tid&31, half = lane>>4, row = lane&15;
  int q = blockIdx.x*128 + wave*16;
  int n = q/P_, p = q - n*P_;
  const float* xb = X + (long long)n*C_*P_ + p + row;
  v8f acc[8];
  #pragma unroll
  for (int t = 0; t < 8; ++t) acc[t] = vzero8();
  for (int k0 = 0; k0 < 64; k0 += 4) {
    int ka = k0 + half*2;
    v2f A; A.x = xb[ka*P_]; A.y = xb[(ka+1)*P_];
    #pragma unroll
    for (int t = 0; t < 8; ++t) {
      int nn = hb*128 + t*16 + row;
      v2f Bv; Bv.x = W1T[ka*256+nn]; Bv.y = W1T[(ka+1)*256+nn];
      acc[t] = wmma4(A, Bv, acc[t]);
    }
  }
  #pragma unroll
  for (int t = 0; t < 8; ++t) {
    int loc = t*16 + row, ch = hb*128 + loc;
    float bs = b1[ch], sm = 0.f, sq = 0.f;
    #pragma unroll
    for (int m = 0; m < 8; ++m) { float v = acc[t][m] + bs; sm += v; sq += v*v; }
    atomicAdd(&ssum[loc], sm); atomicAdd(&ssq[loc], sq);
  }
  __syncthreads();
  if (tid < 128) {
    atomicAdd(&st[hb*128+tid], ssum[tid]);
    atomicAdd(&st[256 + hb*128+tid], ssq[tid]);
  }
}

// fused FFN: recompute t = W1T'*xr + b1, BN affine, exact GELU, LDS transpose,
// f2 = W2T'*g + f2_b ; write f2_raw + its stats
__global__ __launch_bounds__(64) void k_ffn(
    const float* __restrict__ XR, const float* __restrict__ W1T,
    const float* __restrict__ b1, const float* __restrict__ at,
    const float* __restrict__ ct, const float* __restrict__ W2T,
    const float* __restrict__ b2, float* __restrict__ O, float* __restrict__ st) {
  __shared__ float sG[2][16*257];             // [wave][pos][ch], pad 257 vs bank conflicts
  __shared__ float ssum[64], ssq[64];
  int tid = threadIdx.x, wave = tid>>5, lane = tid&31, half = lane>>4, row = lane&15;
  ssum[tid] = 0.f; ssq[tid] = 0.f;
  int q = blockIdx.x*32 + wave*16;
  int n = q/P_, p = q - n*P_;
  const float* xb = XR + (long long)n*C_*P_ + p + row;
  float* g = sG[wave];
  for (int hb = 0; hb < 2; ++hb) {
    v8f acc[8];
    #pragma unroll
    for (int t = 0; t < 8; ++t) acc[t] = vzero8();
    for (int k0 = 0; k0 < 64; k0 += 4) {
      int ka = k0 + half*2;
      v2f A; A.x = xb[ka*P_]; A.y = xb[(ka+1)*P_];
      #pragma unroll
      for (int t = 0; t < 8; ++t) {
        int nn = hb*128 + t*16 + row;
        v2f Bv; Bv.x = W1T[ka*256+nn]; Bv.y = W1T[(ka+1)*256+nn];
        acc[t] = wmma4(A, Bv, acc[t]);
      }
    }
    #pragma unroll
    for (int t = 0; t < 8; ++t) {
      int ch = hb*128 + t*16 + row;
      float aa = at[ch], cc = ct[ch], bb = b1[ch];
      #pragma unroll
      for (int m = 0; m < 8; ++m) {
        float u  = aa*(acc[t][m] + bb) + cc;
        float ge = 0.5f*u*(1.f + erff(u*0.70710678118654752f));   // exact GELU
        g[(half*8 + m)*257 + ch] = ge;
      }
    }
  }
  __syncthreads();
  v8f acc2[4];
  #pragma unroll
  for (int t = 0; t < 4; ++t) acc2[t] = vzero8();
  for (int k0 = 0; k0 < 256; k0 += 4) {
    int ka = k0 + half*2;
    v2f A; A.x = g[row*257 + ka]; A.y = g[row*257 + ka + 1];
    #pragma unroll
    for (int t = 0; t < 4; ++t) {
      int nn = t*16 + row;
      v2f Bv; Bv.x = W2T[ka*64+nn]; Bv.y = W2T[(ka+1)*64+nn];
      acc2[t] = wmma4(A, Bv, acc2[t]);
    }
  }
  #pragma unroll
  for (int t = 0; t < 4; ++t) {
    int ch = t*16 + row;
    float bs = b2[ch], sm = 0.f, sq = 0.f, tmp[8];
    float* op = O + (long long)(n*C_+ch)*P_ + p + half*8;
    #pragma unroll
    for (int m = 0; m < 8; ++m) { float v = acc2[t][m] + bs; tmp[m] = v; sm += v; sq += v*v; }
    *(float4*)op     = make_float4(tmp[0], tmp[1], tmp[2], tmp[3]);
    *(float4*)(op+4) = make_float4(tmp[4], tmp[5], tmp[6], tmp[7]);
    atomicAdd(&ssum[ch], sm); atomicAdd(&ssq[ch], sq);
  }
  __syncthreads();
  atomicAdd(&st[tid], ssum[tid]); atomicAdd(&st[64+tid], ssq[tid]);
}

// =============================== depthwise 13^3 + 5^3 ===============================
__global__ __launch_bounds__(512) void k_dwconv(
    const float* __restrict__ HP, const float* __restrict__ a2, const float* __restrict__ c2,
    const float* __restrict__ w13, const float* __restrict__ b13,
    const float* __restrict__ w5, const float* __restrict__ b5,
    float* __restrict__ Y1, float* __restrict__ Y2,
    float* __restrict__ st1, float* __restrict__ st2) {
  __shared__ float tile[20*20*20];            // 8^3 out + halo 6 (13^3), 32 KB
  __shared__ float red[4];
  int tid = threadIdx.x;
  int tix = blockIdx.x, c = blockIdx.y, n = blockIdx.z;
  int tz = tix/36, ty = (tix/6)%6, tx = tix%6;
  if (tid < 4) red[tid] = 0.f;
  float aa = a2[c], cc = c2[c];
  const float* src = HP + (long long)(n*C_+c)*P_;
  int z0 = tz*8 - 6, y0 = ty*8 - 6, x0 = tx*8 - 6;
  for (int i = tid; i < 8000; i += 512) {     // normalize (pc_bn) while staging; zero-pad
    int iz = i/400, r = i - iz*400, iy = r/20, ix = r - iy*20;
    int gz = z0+iz, gy = y0+iy, gx = x0+ix;
    float v = 0.f;
    if ((unsigned)gz < 48u && (unsigned)gy < 48u && (unsigned)gx < 48u)
      v = aa*src[(gz*48+gy)*48+gx] + cc;
    tile[i] = v;
  }
  __syncthreads();
  int oz = tid>>6, oy = (tid>>3)&7, ox = tid&7;
  const float* wl = w13 + c*2197;
  const float* ws = w5  + c*125;
  float accL = 0.f, accS = 0.f;
  for (int kz = 0; kz < 13; ++kz)
    for (int ky = 0; ky < 13; ++ky) {
      const float* tr = &tile[((oz+kz)*20 + oy+ky)*20 + ox];
      const float* wr = wl + (kz*13+ky)*13;
      #pragma unroll
      for (int kx = 0; kx < 13; ++kx) accL = fmaf(wr[kx], tr[kx], accL);
    }
  for (int kz = 0; kz < 5; ++kz)
    for (int ky = 0; ky < 5; ++ky) {
      const float* tr = &tile[((oz+4+kz)*20 + oy+4+ky)*20 + ox+4];
      const float* wr = ws + (kz*5+ky)*5;
      #pragma unroll
      for (int kx = 0; kx < 5; ++kx) accS = fmaf(wr[kx], tr[kx], accS);
    }
  float y1 = accL + b13[c], y2 = accS + b5[c];
  int z = tz*8+oz, y = ty*8+oy, x = tx*8+ox;
  long long oi = (long long)(n*C_+c)*P_ + (z*48+y)*48 + x;
  Y1[oi] = y1; Y2[oi] = y2;
  atomicAdd(&red[0], y1); atomicAdd(&red[1], y1*y1);
  atomicAdd(&red[2], y2); atomicAdd(&red[3], y2*y2);
  __syncthreads();
  if (tid == 0) {
    atomicAdd(&st1[c], red[0]); atomicAdd(&st1[64+c], red[1]);
    atomicAdd(&st2[c], red[2]); atomicAdd(&st2[64+c], red[3]);
  }
}

// =============================== host-side pipeline ===============================
extern "C" void kernel_launch(void* const* d_in, const int* in_sizes, int n_in,
                              void* d_out, int out_size, void* d_ws, size_t ws_size,
                              hipStream_t stream) {
  const float* x     = (const float*)d_in[0];
  const float* perg  = (const float*)d_in[1];
  const float* perb  = (const float*)d_in[2];
  const float* pcw   = (const float*)d_in[3];
  const float* pcb   = (const float*)d_in[4];
  const float* pcbng = (const float*)d_in[5];
  const float* pcbnb = (const float*)d_in[6];
  const float* lkw   = (const float*)d_in[7];
  const float* lkb   = (const float*)d_in[8];
  const float* lkbng = (const float*)d_in[9];
  // d_in[10] lk_bn_b cancels analytically
  const float* lk2g  = (const float*)d_in[11];
  const float* lk2b  = (const float*)d_in[12];
  const float* skw   = (const float*)d_in[13];
  const float* skb   = (const float*)d_in[14];
  const float* skbng = (const float*)d_in[15];
  // d_in[16] sk_bn_b cancels analytically
  const float* sk2g  = (const float*)d_in[17];
  const float* sk2b  = (const float*)d_in[18];
  const float* poww  = (const float*)d_in[19];
  const float* pob   = (const float*)d_in[20];
  const float* pobng = (const float*)d_in[21];
  const float* pobnb = (const float*)d_in[22];
  const float* ffng  = (const float*)d_in[23];
  const float* ffnb  = (const float*)d_in[24];
  const float* f1w   = (const float*)d_in[25];
  const float* f1b   = (const float*)d_in[26];
  const float* f1bng = (const float*)d_in[27];
  const float* f1bnb = (const float*)d_in[28];
  const float* f2w   = (const float*)d_in[29];
  const float* f2b   = (const float*)d_in[30];
  const float* f2bng = (const float*)d_in[31];
  const float* f2bnb = (const float*)d_in[32];

  float* ws   = (float*)d_ws;
  float* bufA = ws;                 // hp -> po_raw
  float* bufB = ws + NE_;           // y1 -> xr
  float* bufC = ws + 2*NE_;         // y2 -> f2_raw
  float* prm  = ws + 3*NE_;
  float* st_x  = prm + 0;    float* st_hp = prm + 128;
  float* st_y1 = prm + 256;  float* st_y2 = prm + 384;
  float* st_po = prm + 512;  float* st_xr = prm + 640;
  float* st_f2 = prm + 768;  float* st_t  = prm + 896;    // 512 floats
  float* a2  = prm + 1408;   float* c2  = prm + 1472;
  float* apo = prm + 1536;   float* cpo = prm + 1600;
  float* at_ = prm + 1664;   float* ct_ = prm + 1920;
  float* a2f = prm + 2176;   float* c2f = prm + 2240;
  float* biaspc = prm + 2304;
  float* biaspo = prm + 2368;
  float* b1o    = prm + 2432;       // 256
  float* WTpc = prm + 2688;         // 4096
  float* WAT  = prm + 6784;         // 4096
  float* WBT  = prm + 10880;        // 4096
  float* W1T  = prm + 14976;        // 16384
  float* W2T  = prm + 31360;        // 16384

  k_zero<<<(1408+255)/256, 256, 0, stream>>>(prm, 1408);
  k_stats<<<dim3(16,64,2), 256, 0, stream>>>(x, st_x);
  k_fin1<<<1, 256, 0, stream>>>(st_x, perg, perb, pcw, pcb, f2w, WTpc, biaspc, W2T);
  k_gemm64<<<NP_/128, 256, 0, stream>>>(x, WTpc, biaspc, bufA, st_hp);
  k_affine<<<1, 64, 0, stream>>>(st_hp, pcbng, pcbnb, a2, c2, 64);
  k_dwconv<<<dim3(216,64,2), 512, 0, stream>>>(bufA, a2, c2, lkw, lkb, skw, skb,
                                               bufB, bufC, st_y1, st_y2);
  k_fin3<<<1, 256, 0, stream>>>(st_y1, st_y2, lkbng, lk2g, lk2b,
                                skbng, sk2g, sk2b, poww, pob, WAT, WBT, biaspo);
  k_gemm64x2<<<NP_/128, 256, 0, stream>>>(bufB, bufC, WAT, WBT, biaspo, bufA, st_po);
  k_affine<<<1, 64, 0, stream>>>(st_po, pobng, pobnb, apo, cpo, 64);
  k_resid_stats<<<dim3(16,64,2), 256, 0, stream>>>(bufA, x, apo, cpo, bufB, st_xr);
  k_fin5<<<1, 256, 0, stream>>>(st_xr, ffng, ffnb, f1w, f1b, W1T, b1o);
  k_gemm256_stats<<<dim3(NP_/128,2), 256, 0, stream>>>(bufB, W1T, b1o, st_t);
  k_affine<<<1, 256, 0, stream>>>(st_t, f1bng, f1bnb, at_, ct_, 256);
  k_ffn<<<NP_/32, 64, 0, stream>>>(bufB, W1T, b1o, at_, ct_, W2T, f2b, bufC, st_f2);
  k_affine<<<1, 64, 0, stream>>>(st_f2, f2bng, f2bnb, a2f, c2f, 64);
  k_resid_out<<<dim3(16,64,2), 256, 0, stream>>>(bufC, bufB, a2f, c2f, (float*)d_out);
}